// Mamba_71330816852878
// MI455X (gfx1250) — compile-verified
//
#include <hip/hip_runtime.h>

// Mamba forward for MI455X (gfx1250, wave32). fp32 end-to-end, matrix ops on
// V_WMMA_F32_16X16X4_F32 with 2x2 register blocking (32x32 C tile per wave).

#define VV   32000
#define DD   256
#define NLAY 2
#define DI   512
#define DS   512
#define DTRK 16
#define DCV  4
#define LL   512

typedef __attribute__((ext_vector_type(2))) float v2f;
typedef __attribute__((ext_vector_type(8))) float v8f;

__device__ __forceinline__ float silu_f(float v) { return v / (1.f + __expf(-v)); }

#define WMMA_F32(a, b, c) \
  __builtin_amdgcn_wmma_f32_16x16x4_f32(false, (a), false, (b), (short)0, (c), false, false)

// ---------------- embedding gather: x[t,d] = emb[tok[t], d] ----------------
__global__ void mamba_embed(const int* __restrict__ tok, const float* __restrict__ emb,
                            float* __restrict__ x) {
  int t = blockIdx.x, d = threadIdx.x;
  x[t * DD + d] = emb[(size_t)tok[t] * DD + d];
}

// ---------------- RMSNorm over D=256, one block (8 waves) per row ----------
__global__ void mamba_rmsnorm(const float* __restrict__ x, const float* __restrict__ w,
                              const float* __restrict__ b, float* __restrict__ out) {
  int row = blockIdx.x, tid = threadIdx.x;
  float v = x[row * DD + tid];
  float ss = v * v;
#pragma unroll
  for (int off = 16; off >= 1; off >>= 1) ss += __shfl_xor(ss, off, 32);
  __shared__ float red[8];
  if ((tid & 31) == 0) red[tid >> 5] = ss;
  __syncthreads();
  float tot = 0.f;
#pragma unroll
  for (int i = 0; i < 8; ++i) tot += red[i];
  float inv = rsqrtf(tot * (1.f / DD) + 1e-5f);
  out[row * DD + tid] = v * inv * w[tid] + b[tid];
}

// ---------------- WMMA fp32 GEMM: C[M,N] (+)= A[M,K] * W[N,K]^T (+ bias) ---
// One wave computes a 32x32 C tile = 2x2 grid of 16x16 WMMA subtiles, so each
// A / B fragment load feeds two v_wmma ops (1 load : 1 wmma in steady state).
// Fragment layout (ISA 7.12.2): lane = (half = lane>>4, idx = lane&15)
//   A frag: float2 at A[r0+idx][k + 2*half]
//   B frag: float2 at W[c0+idx][k + 2*half]
//   C/D:    vgpr r -> C[r0 + r + 8*half][c0 + idx]
// M must be a multiple of 32; N a multiple of 16 (right edge handled with a
// wave-uniform 2x1 path so EXEC is all-ones around every WMMA).
__global__ void mamba_wmma_gemm_nt(const float* __restrict__ A, int lda,
                                   const float* __restrict__ W, int ldw,
                                   const float* __restrict__ bias,
                                   float* __restrict__ C, int ldc,
                                   int N, int K, int accumulate, int nt_store) {
  const int lane = threadIdx.x & 31;
  const int wave = threadIdx.x >> 5;
  const int tn   = blockIdx.x * 4 + wave;          // 32-wide N tile index
  const int col0 = tn * 32;
  if (col0 >= N) return;                            // wave-uniform exit
  const int row0 = blockIdx.y * 32;
  const int half = lane >> 4;
  const int idx  = lane & 15;
  const bool n1ok = (col0 + 32) <= N;               // second N subtile valid?

  v8f c00 = {}, c01 = {}, c10 = {}, c11 = {};

  auto preload = [&](v8f& c, int r0, int c0) {
#pragma unroll
    for (int r = 0; r < 8; ++r)
      c[r] = C[(size_t)(r0 + r + 8 * half) * ldc + c0 + idx];
  };
  auto emit = [&](const v8f& c, int r0, int c0) {
#pragma unroll
    for (int r = 0; r < 8; ++r) {
      float o = c[r];
      if (bias) o += bias[c0 + idx];
      float* p = &C[(size_t)(r0 + r + 8 * half) * ldc + c0 + idx];
      if (nt_store) __builtin_nontemporal_store(o, p);
      else *p = o;
    }
  };

  if (accumulate) {
    preload(c00, row0, col0);
    preload(c10, row0 + 16, col0);
    if (n1ok) { preload(c01, row0, col0 + 16); preload(c11, row0 + 16, col0 + 16); }
  }

  const float* Ar0 = A + (size_t)(row0 + idx) * lda + 2 * half;
  const float* Ar1 = Ar0 + (size_t)16 * lda;
  const float* Wr0 = W + (size_t)(col0 + idx) * ldw + 2 * half;
  const float* Wr1 = Wr0 + (size_t)16 * ldw;

  if (n1ok) {
    for (int k = 0; k < K; k += 4) {
      v2f a0 = *(const v2f*)(Ar0 + k);
      v2f a1 = *(const v2f*)(Ar1 + k);
      v2f b0 = *(const v2f*)(Wr0 + k);
      v2f b1 = *(const v2f*)(Wr1 + k);
      c00 = WMMA_F32(a0, b0, c00);
      c01 = WMMA_F32(a0, b1, c01);
      c10 = WMMA_F32(a1, b0, c10);
      c11 = WMMA_F32(a1, b1, c11);
    }
    emit(c00, row0, col0);
    emit(c10, row0 + 16, col0);
    emit(c01, row0, col0 + 16);
    emit(c11, row0 + 16, col0 + 16);
  } else {
    for (int k = 0; k < K; k += 4) {
      v2f a0 = *(const v2f*)(Ar0 + k);
      v2f a1 = *(const v2f*)(Ar1 + k);
      v2f b0 = *(const v2f*)(Wr0 + k);
      c00 = WMMA_F32(a0, b0, c00);
      c10 = WMMA_F32(a1, b0, c10);
    }
    emit(c00, row0, col0);
    emit(c10, row0 + 16, col0);
  }
}

// ---------------- causal depthwise conv (DC=4) + SiLU ----------------------
__global__ void mamba_conv_silu(const float* __restrict__ xr, const float* __restrict__ cw,
                                const float* __restrict__ cb, float* __restrict__ xc) {
  int i  = blockIdx.x * blockDim.x + threadIdx.x;   // over L*DI
  int t  = i / DI;
  int ch = i - t * DI;
  float acc = cb[ch];
#pragma unroll
  for (int k = 0; k < DCV; ++k) {
    int tt = t - (DCV - 1) + k;
    float xv = (tt >= 0) ? xr[tt * (2 * DI) + ch] : 0.f;
    acc = fmaf(cw[ch * DCV + k], xv, acc);
  }
  xc[i] = silu_f(acc);
}

// ---------------- softplus in place ----------------------------------------
__global__ void mamba_softplus(float* __restrict__ d) {
  int i = blockIdx.x * blockDim.x + threadIdx.x;
  float v = d[i];
  d[i] = (v > 20.f) ? v : log1pf(__expf(v));
}

// ---------------- selective scan -------------------------------------------
// One wave per channel i (8 channels per 256-thread block). Each lane holds
// 16 states j = s*32+lane in VGPRs. Sequential over t; per step:
//   h = exp(d*A)*h + (d*x)*B_t[j];  y_t[i] = sum_j h*C_t[j] (wave reduce).
__global__ void mamba_scan(const float* __restrict__ xc,    // [L, DI]
                           const float* __restrict__ dbc,   // [L, DTR+2*DS]
                           const float* __restrict__ delta, // [L, DI]
                           const float* __restrict__ A_log, // [DI, DS]
                           float* __restrict__ ys) {        // [L, DI]
  const int lane = threadIdx.x & 31;
  const int wave = threadIdx.x >> 5;
  const int ch   = blockIdx.x * 8 + wave;

  float Arow[16], h[16];
#pragma unroll
  for (int s = 0; s < 16; ++s) {
    Arow[s] = -__expf(A_log[(size_t)ch * DS + s * 32 + lane]);
    h[s] = 0.f;
  }

  const int LDBC = DTRK + 2 * DS;
  for (int t = 0; t < LL; ++t) {
    float d  = delta[t * DI + ch];
    float xv = xc[t * DI + ch];
    float dx = d * xv;
    const float* Bt = dbc + (size_t)t * LDBC + DTRK;
    const float* Ct = Bt + DS;
    float acc = 0.f;
#pragma unroll
    for (int s = 0; s < 16; ++s) {
      int j = s * 32 + lane;
      float dA = __expf(d * Arow[s]);
      h[s] = fmaf(dA, h[s], dx * Bt[j]);
      acc  = fmaf(h[s], Ct[j], acc);
    }
#pragma unroll
    for (int off = 16; off >= 1; off >>= 1) acc += __shfl_xor(acc, off, 32);
    if (lane == 0) ys[t * DI + ch] = acc;
  }
}

// ---------------- gate: g = (ys + xc*Dp) * silu(res) -----------------------
__global__ void mamba_gate(const float* __restrict__ ys, const float* __restrict__ xc,
                           const float* __restrict__ Dp, const float* __restrict__ xr,
                           float* __restrict__ g) {
  int i  = blockIdx.x * blockDim.x + threadIdx.x;
  int t  = i / DI;
  int ch = i - t * DI;
  float y = fmaf(xc[i], Dp[ch], ys[i]);
  float r = xr[t * (2 * DI) + DI + ch];
  g[i] = y * silu_f(r);
}

// ---------------------------------------------------------------------------
extern "C" void kernel_launch(void* const* d_in, const int* in_sizes, int n_in,
                              void* d_out, int out_size, void* d_ws, size_t ws_size,
                              hipStream_t stream) {
  (void)in_sizes; (void)n_in; (void)out_size; (void)ws_size;

  const int*   tokens  = (const int*)  d_in[0];
  const float* emb_W   = (const float*)d_in[1];
  const float* norm_w  = (const float*)d_in[2];
  const float* norm_b  = (const float*)d_in[3];
  const float* in_W    = (const float*)d_in[4];
  const float* in_b    = (const float*)d_in[5];
  const float* conv_W  = (const float*)d_in[6];
  const float* conv_b  = (const float*)d_in[7];
  const float* xproj_W = (const float*)d_in[8];
  const float* dt_W    = (const float*)d_in[9];
  const float* A_log   = (const float*)d_in[10];
  const float* D_p     = (const float*)d_in[11];
  const float* out_W   = (const float*)d_in[12];
  const float* out_b   = (const float*)d_in[13];
  const float* fnorm_w = (const float*)d_in[14];
  const float* fnorm_b = (const float*)d_in[15];
  float* logits = (float*)d_out;

  const int LDBC = DTRK + 2 * DS;   // 1040
  float* ws  = (float*)d_ws;
  float* x   = ws;                  // [L, D]
  float* h   = x   + LL * DD;       // [L, D]
  float* xr  = h   + LL * DD;       // [L, 2*DI]
  float* xcb = xr  + LL * 2 * DI;   // [L, DI]
  float* dbc = xcb + LL * DI;       // [L, 1040]
  float* dlt = dbc + LL * LDBC;     // [L, DI]
  float* ysb = dlt + LL * DI;       // [L, DI]
  float* gat = ysb + LL * DI;       // [L, DI]

  const dim3 blk256(256), blkG(128);
  const int  MT32 = LL / 32;        // 16 row tiles (32-wide) everywhere
  auto ntiles = [](int N) { return ((N + 31) / 32 + 3) / 4; };  // blocks.x (4 waves/blk)

  mamba_embed<<<dim3(LL), blk256, 0, stream>>>(tokens, emb_W, x);

  for (int l = 0; l < NLAY; ++l) {
    // h = rmsnorm(x)
    mamba_rmsnorm<<<dim3(LL), blk256, 0, stream>>>(x, norm_w + l * DD, norm_b + l * DD, h);
    // xr = h @ in_W^T + in_b            [512,256]x[1024,256]^T
    mamba_wmma_gemm_nt<<<dim3(ntiles(2 * DI), MT32), blkG, 0, stream>>>(
        h, DD, in_W + (size_t)l * 2 * DI * DD, DD, in_b + l * 2 * DI,
        xr, 2 * DI, 2 * DI, DD, 0, 0);
    // xc = silu(causal depthwise conv(xr[:, :DI]))
    mamba_conv_silu<<<dim3(LL * DI / 256), blk256, 0, stream>>>(
        xr, conv_W + (size_t)l * DI * DCV, conv_b + l * DI, xcb);
    // dbc = xc @ xproj_W^T              [512,512]x[1040,512]^T (N=1040: ragged edge)
    mamba_wmma_gemm_nt<<<dim3(ntiles(LDBC), MT32), blkG, 0, stream>>>(
        xcb, DI, xproj_W + (size_t)l * LDBC * DI, DI, nullptr,
        dbc, LDBC, LDBC, DI, 0, 0);
    // dlt = dbc[:, :16] @ dt_W^T        [512,16]x[512,16]^T, then softplus
    mamba_wmma_gemm_nt<<<dim3(ntiles(DI), MT32), blkG, 0, stream>>>(
        dbc, LDBC, dt_W + (size_t)l * DI * DTRK, DTRK, nullptr,
        dlt, DI, DI, DTRK, 0, 0);
    mamba_softplus<<<dim3(LL * DI / 256), blk256, 0, stream>>>(dlt);
    // selective scan
    mamba_scan<<<dim3(DI / 8), blk256, 0, stream>>>(
        xcb, dbc, dlt, A_log + (size_t)l * DI * DS, ysb);
    // gate with residual branch
    mamba_gate<<<dim3(LL * DI / 256), blk256, 0, stream>>>(
        ysb, xcb, D_p + l * DI, xr, gat);
    // x += gat @ out_W^T + out_b        [512,512]x[256,512]^T, accumulate
    mamba_wmma_gemm_nt<<<dim3(ntiles(DD), MT32), blkG, 0, stream>>>(
        gat, DI, out_W + (size_t)l * DD * DI, DI, out_b + l * DD,
        x, DD, DD, DI, 1, 0);
  }

  // final norm + tied LM head: logits = rms(x) @ emb_W^T  [512,256]x[32000,256]^T
  // Logits are write-once (65.5 MB) -> non-temporal stores keep L2 for weights.
  mamba_rmsnorm<<<dim3(LL), blk256, 0, stream>>>(x, fnorm_w, fnorm_b, h);
  mamba_wmma_gemm_nt<<<dim3(ntiles(VV), MT32), blkG, 0, stream>>>(
      h, DD, emb_W, DD, nullptr, logits, VV, VV, DD, 0, 1);
}